// AngularDescriptor_39728447488458
// MI455X (gfx1250) — compile-verified
//
#include <hip/hip_runtime.h>

typedef __attribute__((ext_vector_type(2))) float v2f;
typedef __attribute__((ext_vector_type(8))) float v8f;

#define N_TYPES 4
#define N_DESC  8
#define K_MAX   8
#define MAX_NEI 64
#define R_C     4.0f
#define PI_OVER_RC 0.78539816339744830962f
#define ATOMS_PER_BLOCK 4
#define THREADS (ATOMS_PER_BLOCK * 32)

__device__ __constant__ float cC3B[24] = {
    0.238732414637843f, 0.119366207318922f, 0.119366207318922f,
    0.099471839432435f, 0.596831036594608f, 0.596831036594608f,
    0.149207759148652f, 0.149207759148652f,
    0.139260575205408f, 0.104445431404056f, 0.104445431404056f,
    1.044454314040563f, 1.044454314040563f, 0.174075719006761f,
    0.174075719006761f,
    0.011190581936149f, 0.223811638722978f, 0.223811638722978f,
    0.111905819361489f, 0.111905819361489f, 1.566681471060845f,
    1.566681471060845f, 0.195835183882606f, 0.195835183882606f};
__device__ __constant__ float cC4B[5] = {
    -0.007499480826664f, -0.134990654879954f, 0.067495327439977f,
    0.404971964639861f, -0.809943929279723f};
__device__ __constant__ float cC5B[3] = {
    0.026596810706114f, 0.053193621412227f, 0.026596810706114f};

__global__ void __launch_bounds__(THREADS)
angular_desc_kernel(const int* __restrict__ types,
                    const float* __restrict__ positions,
                    const int* __restrict__ neighbors,
                    const float* __restrict__ c_table,
                    float* __restrict__ out, int nAtoms) {
    __shared__ float s_ctab[N_TYPES * N_TYPES * N_DESC * K_MAX];   // 1024 f
    __shared__ float s_gn [ATOMS_PER_BLOCK][32][17];               // nei x desc(pad 16)+1
    __shared__ float s_blm[ATOMS_PER_BLOCK][32][33];               // nei x comp(pad 32)+1
    __shared__ float s_sv [ATOMS_PER_BLOCK][N_DESC][24];

    const int w    = threadIdx.x >> 5;
    const int lane = threadIdx.x & 31;
    const int row  = lane & 15;
    const bool hi  = lane >= 16;

    // stage the 4 KB coefficient table once per block
    for (int i = threadIdx.x; i < N_TYPES * N_TYPES * N_DESC * K_MAX; i += THREADS)
        s_ctab[i] = c_table[i];
    __syncthreads();

    int atom = blockIdx.x * ATOMS_PER_BLOCK + w;
    if (atom >= nAtoms) atom = nAtoms - 1;   // duplicate work, keeps barriers uniform

    const int   ti  = types[atom];
    const float pix = positions[atom * 3 + 0];
    const float piy = positions[atom * 3 + 1];
    const float piz = positions[atom * 3 + 2];

    v8f acc0 = {};  // s[d][c], c = 0..15
    v8f acc1 = {};  // s[d][c], c = 16..23 (24..31 padded zero)

    #pragma unroll
    for (int stage = 0; stage < 2; ++stage) {
        // ---- per-lane: one neighbor's full per-pair chain -----------------
        const int j  = stage * 32 + lane;
        const int nb = neighbors[atom * MAX_NEI + j];
        const bool valid = nb >= 0;
        const int idx = valid ? nb : 0;

        const float rx = positions[idx * 3 + 0] - pix;
        const float ry = positions[idx * 3 + 1] - piy;
        const float rz = positions[idx * 3 + 2] - piz;
        const float d2 = rx * rx + ry * ry + rz * rz;
        const bool safe = d2 > 1e-16f;
        const bool msk  = valid && safe;
        const float dd  = sqrtf(safe ? d2 : 1.0f);

        // Chebyshev radial basis
        const float fc = (dd < R_C) ? (0.5f * cosf(PI_OVER_RC * dd) + 0.5f) : 0.0f;
        const float t  = dd / R_C - 1.0f;
        const float xc = 2.0f * t * t - 1.0f;
        float T[K_MAX];
        T[0] = 1.0f; T[1] = xc;
        #pragma unroll
        for (int k = 2; k < K_MAX; ++k) T[k] = 2.0f * xc * T[k - 1] - T[k - 2];
        float fnb[K_MAX];
        const float hfc = 0.5f * fc;
        #pragma unroll
        for (int k = 0; k < K_MAX; ++k) fnb[k] = (T[k] + 1.0f) * hfc;

        // gn[d] = c_table[ti,tj,d,:] . fn  (masked), zero-pad rows 8..15
        const int tj = types[idx];
        const int cbase = (ti * N_TYPES + tj) * N_DESC * K_MAX;
        #pragma unroll
        for (int d = 0; d < N_DESC; ++d) {
            float g = 0.0f;
            #pragma unroll
            for (int k = 0; k < K_MAX; ++k) g += s_ctab[cbase + d * K_MAX + k] * fnb[k];
            s_gn[w][lane][d] = msk ? g : 0.0f;
        }
        #pragma unroll
        for (int d = N_DESC; d < 16; ++d) s_gn[w][lane][d] = 0.0f;

        // blm components from unit vector (24), zero-pad 24..31
        const float inv_d = safe ? (1.0f / dd) : 0.0f;
        const float x = rx * inv_d, y = ry * inv_d, z = rz * inv_d;
        const float z2 = z * z, z3 = z2 * z, z4 = z2 * z2;
        const float rxx = x * x - y * y, rxy = 2.0f * x * y;          // (x+iy)^2
        const float r3 = x * rxx - y * rxy, i3 = x * rxy + y * rxx;   // (x+iy)^3
        const float r4 = x * r3 - y * i3,  i4 = x * i3 + y * r3;      // (x+iy)^4
        float b[24];
        b[0] = z;  b[1] = x;  b[2] = y;
        b[3] = 3.0f * z2 - 1.0f;  b[4] = z * x;  b[5] = z * y;
        b[6] = rxx;  b[7] = rxy;
        b[8] = 5.0f * z3 - 3.0f * z;
        const float zf9 = 5.0f * z2 - 1.0f;
        b[9] = zf9 * x;  b[10] = zf9 * y;
        b[11] = z * rxx; b[12] = z * rxy;
        b[13] = r3;      b[14] = i3;
        b[15] = 35.0f * z4 - 30.0f * z2 + 3.0f;
        const float zf16 = 7.0f * z3 - 3.0f * z;
        b[16] = zf16 * x; b[17] = zf16 * y;
        const float zf18 = 7.0f * z2 - 1.0f;
        b[18] = zf18 * rxx; b[19] = zf18 * rxy;
        b[20] = z * r3;  b[21] = z * i3;
        b[22] = r4;      b[23] = i4;
        #pragma unroll
        for (int c = 0; c < 24; ++c) s_blm[w][lane][c] = b[c];
        #pragma unroll
        for (int c = 24; c < 32; ++c) s_blm[w][lane][c] = 0.0f;

        __syncthreads();

        // ---- WMMA: A = gn (16x4), B = blm (4x16), K = 4 neighbors/step ----
        // ISA layout: lanes 0-15 hold K=0,1 ; lanes 16-31 hold K=2,3
        #pragma unroll
        for (int c4 = 0; c4 < 8; ++c4) {
            const int n0 = c4 * 4 + (hi ? 2 : 0);
            const int n1 = n0 + 1;
            v2f a, b0, b1;
            a[0]  = s_gn[w][n0][row];        a[1]  = s_gn[w][n1][row];
            b0[0] = s_blm[w][n0][row];       b0[1] = s_blm[w][n1][row];
            b1[0] = s_blm[w][n0][16 + row];  b1[1] = s_blm[w][n1][16 + row];
            acc0 = __builtin_amdgcn_wmma_f32_16x16x4_f32(
                false, a, false, b0, (short)0, acc0, false, false);
            acc1 = __builtin_amdgcn_wmma_f32_16x16x4_f32(
                false, a, false, b1, (short)0, acc1, false, false);
        }
        __syncthreads();   // before next stage overwrites LDS
    }

    // ---- spill s[d][c] (C/D layout: VGPR r, lanes 0-15 -> M=r, N=lane) ----
    if (lane < 16) {
        #pragma unroll
        for (int r = 0; r < N_DESC; ++r) s_sv[w][r][lane] = acc0[r];
        if (lane < 8) {
            #pragma unroll
            for (int r = 0; r < N_DESC; ++r) s_sv[w][r][16 + lane] = acc1[r];
        }
    }
    __syncthreads();

    // ---- per-descriptor polynomial outputs (6 per descriptor) ------------
    if (lane < N_DESC) {
        float s[24];
        #pragma unroll
        for (int c = 0; c < 24; ++c) s[c] = s_sv[w][lane][c];

        float q[6];
        #pragma unroll
        for (int L = 1; L <= 4; ++L) {
            const int st = L * L - 1;
            float a = cC3B[st] * s[st] * s[st];
            #pragma unroll
            for (int i = 1; i < 2 * L + 1; ++i)
                a += 2.0f * cC3B[st + i] * s[st + i] * s[st + i];
            q[L - 1] = a;
        }
        const float s3 = s[3], s4 = s[4], s5 = s[5], s6 = s[6], s7 = s[7];
        q[4] = cC4B[0] * s3 * s3 * s3
             + cC4B[1] * s3 * (s4 * s4 + s5 * s5)
             + cC4B[2] * s3 * (s6 * s6 + s7 * s7)
             + cC4B[3] * s6 * (s5 * s5 - s4 * s4)
             + cC4B[4] * s4 * s5 * s7;
        const float s0q = s[0] * s[0];
        const float s12 = s[1] * s[1] + s[2] * s[2];
        q[5] = cC5B[0] * s0q * s0q + cC5B[1] * s0q * s12 + cC5B[2] * s12 * s12;

        #pragma unroll
        for (int k = 0; k < 6; ++k)
            out[atom * (N_DESC * 6) + lane * 6 + k] = q[k];
    }
}

extern "C" void kernel_launch(void* const* d_in, const int* in_sizes, int n_in,
                              void* d_out, int out_size, void* d_ws, size_t ws_size,
                              hipStream_t stream) {
    const int*   types     = (const int*)d_in[0];
    const float* positions = (const float*)d_in[1];
    const int*   neighbors = (const int*)d_in[2];
    const float* c_table   = (const float*)d_in[3];
    float*       out       = (float*)d_out;

    const int nAtoms = in_sizes[0];
    const int blocks = (nAtoms + ATOMS_PER_BLOCK - 1) / ATOMS_PER_BLOCK;
    angular_desc_kernel<<<blocks, THREADS, 0, stream>>>(
        types, positions, neighbors, c_table, out, nAtoms);
}